// HGTTemporalTransformer_19885698581033
// MI455X (gfx1250) — compile-verified
//
#include <hip/hip_runtime.h>
#include <math.h>

// ---------------- problem constants (from reference) ----------------
#define NN   65536      // N = NB*T
#define FD   64         // node feature dim
#define HD   128        // hidden
#define EE   30000      // edges
#define NTY  5
#define ETY  10
#define NLS  3          // spatial layers
#define NTL  2          // temporal layers
#define NHD  4          // heads
#define DHD  32         // head dim
#define FFD  512        // FFN dim
#define NBB  1024       // N_base
#define TTS  64         // timesteps
#define NGRP (ETY*NTY)                 // 50 edge groups
#define CHK  256
#define NCHK ((EE + CHK - 1)/CHK)      // 118
#define MAXTILES (EE/16 + NGRP)        // 1925 upper bound on q tiles
#define EB64 ((EE + 63)/64)            // 469

typedef __attribute__((ext_vector_type(2))) float v2f;
typedef __attribute__((ext_vector_type(8))) float v8f;

__device__ __forceinline__ v8f wmma_f32_16x16x4(v2f a, v2f b, v8f c) {
  // D = A(16x4,f32) x B(4x16,f32) + C(16x16,f32); wave32
  return __builtin_amdgcn_wmma_f32_16x16x4_f32(false, a, false, b, (short)0, c, false, false);
}

__device__ __forceinline__ float gelu_exact(float x) {
  return 0.5f * x * (1.0f + erff(x * 0.70710678118654752440f));
}

__device__ __forceinline__ float wave_sum32(float v) {
  #pragma unroll
  for (int off = 16; off > 0; off >>= 1) v += __shfl_xor(v, off, 32);
  return v;
}

// ======================= edge grouping (deterministic, stable) =======================
__global__ void edge_hist(const int* __restrict__ ei, const int* __restrict__ etyp,
                          const int* __restrict__ ntyp, int* __restrict__ hist) {
  __shared__ int lh[NGRP];
  int c = blockIdx.x, t = threadIdx.x;
  if (t < NGRP) lh[t] = 0;
  __syncthreads();
  int e = c * CHK + t;
  if (e < EE) {
    int g = etyp[e] * NTY + ntyp[ei[EE + e]];   // (edge_type, dst node type)
    atomicAdd(&lh[g], 1);
  }
  __syncthreads();
  if (t < NGRP) hist[c * NGRP + t] = lh[t];
}

__global__ void edge_scan(const int* __restrict__ hist, int* __restrict__ gstart,
                          int* __restrict__ goff) {
  int tot[NGRP];
  for (int g = 0; g < NGRP; g++) tot[g] = 0;
  for (int c = 0; c < NCHK; c++)
    for (int g = 0; g < NGRP; g++) tot[g] += hist[c * NGRP + g];
  goff[0] = 0;
  for (int g = 0; g < NGRP; g++) goff[g + 1] = goff[g] + tot[g];
  int cur[NGRP];
  for (int g = 0; g < NGRP; g++) cur[g] = goff[g];
  for (int c = 0; c < NCHK; c++)
    for (int g = 0; g < NGRP; g++) { gstart[c * NGRP + g] = cur[g]; cur[g] += hist[c * NGRP + g]; }
}

__global__ void edge_place(const int* __restrict__ ei, const int* __restrict__ etyp,
                           const int* __restrict__ ntyp, const int* __restrict__ gstart,
                           int* __restrict__ eperm) {
  int c = blockIdx.x;
  int cur[NGRP];
  for (int g = 0; g < NGRP; g++) cur[g] = gstart[c * NGRP + g];
  for (int i = 0; i < CHK; i++) {
    int e = c * CHK + i;
    if (e < EE) {
      int g = etyp[e] * NTY + ntyp[ei[EE + e]];
      eperm[cur[g]++] = e;
    }
  }
}

__global__ void build_idx(const int* __restrict__ ei, const int* __restrict__ eperm,
                          int* __restrict__ idxQ, int* __restrict__ idxKV) {
  int e = blockIdx.x * 256 + threadIdx.x;
  if (e < EE) {
    int oe = eperm[e];
    idxQ[e]  = ei[EE + oe];   // dst node (query side, also scatter target)
    idxKV[e] = ei[oe];        // src node
  }
}

__global__ void build_tiles(const int* __restrict__ goff, int* __restrict__ tileg,
                            int* __restrict__ tileq, int* __restrict__ ntt) {
  int idx = 0;
  for (int g = 0; g < NGRP; g++) {
    int c0 = goff[g], c1 = goff[g + 1];
    for (int q = c0; q < c1; q += 16) { tileg[idx] = g; tileq[idx] = q; idx++; }
  }
  *ntt = idx;
}

// ======================= simple elementwise kernels =======================
__global__ void zero_f32(float* __restrict__ p, int n) {
  int i = blockIdx.x * 256 + threadIdx.x;
  if (i < n) p[i] = 0.f;
}

__global__ void input_proj(const float* __restrict__ x, const int* __restrict__ ntyp,
                           const float* __restrict__ Win, const float* __restrict__ bin,
                           float* __restrict__ h) {
  int n = blockIdx.x, j = threadIdx.x;
  int nt = ntyp[n];
  const float* xr = x + (size_t)n * FD;
  const float* wr = Win + (size_t)nt * FD * HD;
  float acc = bin[nt * HD + j];
  #pragma unroll 8
  for (int f = 0; f < FD; f++) acc += xr[f] * wr[f * HD + j];
  h[(size_t)n * HD + j] = acc;
}

__global__ void pe_add(float* __restrict__ h) {
  int n = blockIdx.x, j = threadIdx.x;
  int t = n & (TTS - 1);
  float d = expf((float)(j & ~1) * (-0.0719557841560639f)); // -ln(10000)/128 * (2i)
  float v = (j & 1) ? cosf((float)t * d) : sinf((float)t * d);
  h[(size_t)n * HD + j] += v;
}

__global__ void scatter_add(const float* __restrict__ eout, const int* __restrict__ dstnode,
                            float* __restrict__ hnew) {
  int e = blockIdx.x, t = threadIdx.x;
  atomicAdd(&hnew[(size_t)dstnode[e] * HD + t], eout[(size_t)e * HD + t]);
}

// h = (addres ? h : 0) + gelu(LN(src))
__global__ void ln_gelu_res(float* __restrict__ h, const float* __restrict__ src,
                            const float* __restrict__ g, const float* __restrict__ b, int addres) {
  int row = blockIdx.x * 4 + (threadIdx.x >> 5);
  int lane = threadIdx.x & 31;
  const float* s = src + (size_t)row * HD;
  float v[4];
  float sum = 0.f;
  #pragma unroll
  for (int i = 0; i < 4; i++) { v[i] = s[lane + 32 * i]; sum += v[i]; }
  float mean = wave_sum32(sum) * (1.f / HD);
  float var = 0.f;
  #pragma unroll
  for (int i = 0; i < 4; i++) { float d = v[i] - mean; var += d * d; }
  var = wave_sum32(var) * (1.f / HD);
  float inv = rsqrtf(var + 1e-5f);
  float* hr = h + (size_t)row * HD;
  #pragma unroll
  for (int i = 0; i < 4; i++) {
    int c = lane + 32 * i;
    float y = gelu_exact((v[i] - mean) * inv * g[c] + b[c]);
    hr[c] = addres ? (hr[c] + y) : y;
  }
}

// h = LN(h + delta)
__global__ void add_ln(float* __restrict__ h, const float* __restrict__ delta,
                       const float* __restrict__ g, const float* __restrict__ b) {
  int row = blockIdx.x * 4 + (threadIdx.x >> 5);
  int lane = threadIdx.x & 31;
  float* hr = h + (size_t)row * HD;
  const float* dr = delta + (size_t)row * HD;
  float v[4];
  float sum = 0.f;
  #pragma unroll
  for (int i = 0; i < 4; i++) { v[i] = hr[lane + 32 * i] + dr[lane + 32 * i]; sum += v[i]; }
  float mean = wave_sum32(sum) * (1.f / HD);
  float var = 0.f;
  #pragma unroll
  for (int i = 0; i < 4; i++) { float d = v[i] - mean; var += d * d; }
  var = wave_sum32(var) * (1.f / HD);
  float inv = rsqrtf(var + 1e-5f);
  #pragma unroll
  for (int i = 0; i < 4; i++) {
    int c = lane + 32 * i;
    hr[c] = (v[i] - mean) * inv * g[c] + b[c];
  }
}

__global__ void final_ln(const float* __restrict__ h, const float* __restrict__ g,
                         const float* __restrict__ b, float* __restrict__ out) {
  int row = blockIdx.x * 4 + (threadIdx.x >> 5);
  int lane = threadIdx.x & 31;
  const float* hr = h + (size_t)row * HD;
  float v[4];
  float sum = 0.f;
  #pragma unroll
  for (int i = 0; i < 4; i++) { v[i] = hr[lane + 32 * i]; sum += v[i]; }
  float mean = wave_sum32(sum) * (1.f / HD);
  float var = 0.f;
  #pragma unroll
  for (int i = 0; i < 4; i++) { float d = v[i] - mean; var += d * d; }
  var = wave_sum32(var) * (1.f / HD);
  float inv = rsqrtf(var + 1e-5f);
  #pragma unroll
  for (int i = 0; i < 4; i++) {
    int c = lane + 32 * i;
    out[(size_t)row * HD + c] = (v[i] - mean) * inv * g[c] + b[c];
  }
}

// ======================= WMMA GEMM core: C[64x64] = A @ W^T + bias =======================
// A rows: global row rg = rowbase + r (r<64); gathered through rowidx if non-null.
// Rows with rg >= Mend are compute-garbage but never stored. W: [64 rows x K] (caller
// pre-offsets W/bias/C to the n0 column block). blockDim must be 128 (4 waves, 2x2).
__device__ __forceinline__ void gemm_core(const float* __restrict__ A, int lda,
                                          const int* __restrict__ rowidx,
                                          int rowbase, int Mend,
                                          const float* __restrict__ W, int K,
                                          const float* __restrict__ bias,
                                          float* __restrict__ C, int ldc, int act) {
  __shared__ float As[64][17];
  __shared__ float Ws[64][17];
  int tid = threadIdx.x;
  int w = tid >> 5, lane = tid & 31, ll = lane & 15, hl = lane >> 4;
  int wm = (w >> 1) * 32, wn = (w & 1) * 32;
  v8f a00 = {}, a01 = {}, a10 = {}, a11 = {};
  for (int k0 = 0; k0 < K; k0 += 16) {
    __syncthreads();
    for (int i = tid; i < 64 * 16; i += 128) {
      int r = i >> 4, kk = i & 15;
      int rg = rowbase + r; if (rg > Mend - 1) rg = Mend - 1;
      int ar = rowidx ? rowidx[rg] : rg;
      As[r][kk] = A[(size_t)ar * lda + k0 + kk];
      Ws[r][kk] = W[(size_t)r * K + k0 + kk];
    }
    __syncthreads();
    #pragma unroll
    for (int kk = 0; kk < 16; kk += 4) {
      v2f fa0 = { As[wm + ll][kk + 2 * hl],      As[wm + ll][kk + 2 * hl + 1] };
      v2f fa1 = { As[wm + 16 + ll][kk + 2 * hl], As[wm + 16 + ll][kk + 2 * hl + 1] };
      v2f fb0 = { Ws[wn + ll][kk + 2 * hl],      Ws[wn + ll][kk + 2 * hl + 1] };
      v2f fb1 = { Ws[wn + 16 + ll][kk + 2 * hl], Ws[wn + 16 + ll][kk + 2 * hl + 1] };
      a00 = wmma_f32_16x16x4(fa0, fb0, a00);
      a01 = wmma_f32_16x16x4(fa0, fb1, a01);
      a10 = wmma_f32_16x16x4(fa1, fb0, a10);
      a11 = wmma_f32_16x16x4(fa1, fb1, a11);
    }
  }
  // store (C layout: VGPR r -> M=r (lanes 0-15) / M=r+8 (lanes 16-31), N=ll)
  auto store_tile = [&](v8f acc, int mt, int nt) {
    int col = nt + ll;
    float bb = bias[col];
    #pragma unroll
    for (int r = 0; r < 8; r++) {
      int rg = rowbase + mt + r + hl * 8;
      if (rg < Mend) {
        float y = acc[r] + bb;
        if (act) y = gelu_exact(y);
        C[(size_t)rg * ldc + col] = y;
      }
    }
  };
  store_tile(a00, wm, wn);
  store_tile(a01, wm, wn + 16);
  store_tile(a10, wm + 16, wn);
  store_tile(a11, wm + 16, wn + 16);
}

__global__ void __launch_bounds__(128)
gemm_plain(const float* __restrict__ A, int lda, int M,
           const float* __restrict__ W, int K, const float* __restrict__ bias,
           float* __restrict__ C, int ldc, int act) {
  int m0 = blockIdx.x * 64;
  if (m0 >= M) return;
  int n0 = blockIdx.y * 64;
  gemm_core(A, lda, nullptr, m0, M, W + (size_t)n0 * K, K, bias + n0, C + n0, ldc, act);
}

// per-edge-type QKV projection (gather rows of h by dst (Q cols) / src (K,V cols))
__global__ void __launch_bounds__(128)
spatial_qkv(const float* __restrict__ h, const int* __restrict__ idxQ,
            const int* __restrict__ idxKV, const int* __restrict__ goff, int et,
            const float* __restrict__ Wi, const float* __restrict__ bi,
            float* __restrict__ QKV) {
  int base = goff[et * NTY], end = goff[et * NTY + NTY];
  int m0 = base + blockIdx.x * 64;
  if (m0 >= end) return;
  int n0 = blockIdx.y * 64;                       // 0..320 : Q=[0,128) K=[128,256) V=[256,384)
  const int* idx = (n0 < HD) ? idxQ : idxKV;
  gemm_core(h, HD, idx, m0, end, Wi + (size_t)n0 * HD, HD, bi + n0, QKV + n0, 3 * HD, 0);
}

__global__ void __launch_bounds__(128)
spatial_outproj(const float* __restrict__ Ain, const int* __restrict__ goff, int et,
                const float* __restrict__ W, const float* __restrict__ bias,
                float* __restrict__ C) {
  int base = goff[et * NTY], end = goff[et * NTY + NTY];
  int m0 = base + blockIdx.x * 64;
  if (m0 >= end) return;
  int n0 = blockIdx.y * 64;
  gemm_core(Ain, HD, nullptr, m0, end, W + (size_t)n0 * HD, HD, bias + n0, C + n0, HD, 0);
}

// ======================= spatial grouped attention (flash style) =======================
// One block per 16-row q-tile of a group; one wave per head; online softmax over k tiles.
__global__ void __launch_bounds__(128)
spatial_attn(const float* __restrict__ QKV, const int* __restrict__ goff,
             const int* __restrict__ tileg, const int* __restrict__ tileq,
             const int* __restrict__ ntt, float* __restrict__ Oe) {
  if ((int)blockIdx.x >= *ntt) return;
  int g = tileg[blockIdx.x];
  int q0 = tileq[blockIdx.x];
  int gbeg = goff[g], gend = goff[g + 1];
  __shared__ float qs[16][132];
  __shared__ float ks[16][132];
  __shared__ float vs[16][132];
  __shared__ float ps[NHD][16][18];
  __shared__ float rmax[NHD][16], rsum[NHD][16], ralpha[NHD][16];
  int tid = threadIdx.x, w = tid >> 5, lane = tid & 31, ll = lane & 15, hl = lane >> 4;
  for (int i = tid; i < 16 * HD; i += 128) {
    int r = i >> 7, c = i & 127;
    int e = q0 + r;
    qs[r][c] = (e < gend) ? QKV[(size_t)e * (3 * HD) + c] : 0.f;
  }
  if (tid < NHD * 16) {
    int hh = tid >> 4, rr = tid & 15;
    rmax[hh][rr] = -1e30f; rsum[hh][rr] = 0.f;
  }
  v8f o0 = {}, o1 = {};
  int nk = (gend - gbeg + 15) >> 4;
  const float scale = 0.17677669529663687f;   // 1/sqrt(32)
  for (int kt = 0; kt < nk; kt++) {
    int kb = gbeg + kt * 16;
    __syncthreads();
    for (int i = tid; i < 16 * HD; i += 128) {
      int r = i >> 7, c = i & 127;
      int e = kb + r;
      float kv = (e < gend) ? QKV[(size_t)e * (3 * HD) + HD + c] : 0.f;
      float vv = (e < gend) ? QKV[(size_t)e * (3 * HD) + 2 * HD + c] : 0.f;
      ks[r][c] = kv;
      vs[r][c] = vv;
    }
    __syncthreads();
    // S_h = Q_h @ K_h^T  (wave w == head)
    v8f sc = {};
    #pragma unroll
    for (int kk = 0; kk < DHD; kk += 4) {
      v2f a = { qs[ll][w * DHD + kk + 2 * hl], qs[ll][w * DHD + kk + 2 * hl + 1] };
      v2f b = { ks[ll][w * DHD + kk + 2 * hl], ks[ll][w * DHD + kk + 2 * hl + 1] };
      sc = wmma_f32_16x16x4(a, b, sc);
    }
    #pragma unroll
    for (int r = 0; r < 8; r++) ps[w][r + hl * 8][ll] = sc[r] * scale;
    __syncthreads();
    // online softmax update (lanes 0..15 of each wave handle 16 q rows of its head)
    if (lane < 16) {
      int rr = lane;
      int nv = gend - kb; if (nv > 16) nv = 16;
      float mx = rmax[w][rr];
      for (int c = 0; c < nv; c++) mx = fmaxf(mx, ps[w][rr][c]);
      float al = expf(rmax[w][rr] - mx);
      float sm = rsum[w][rr] * al;
      for (int c = 0; c < 16; c++) {
        float p = (c < nv) ? expf(ps[w][rr][c] - mx) : 0.f;
        ps[w][rr][c] = p;
        sm += p;
      }
      rmax[w][rr] = mx; rsum[w][rr] = sm; ralpha[w][rr] = al;
    }
    __syncthreads();
    #pragma unroll
    for (int r = 0; r < 8; r++) {
      float al = ralpha[w][r + hl * 8];
      o0[r] *= al; o1[r] *= al;
    }
    // O_h += P @ V_h
    #pragma unroll
    for (int kk = 0; kk < 16; kk += 4) {
      v2f a  = { ps[w][ll][kk + 2 * hl], ps[w][ll][kk + 2 * hl + 1] };
      v2f b0 = { vs[kk + 2 * hl][w * DHD + ll],      vs[kk + 2 * hl + 1][w * DHD + ll] };
      v2f b1 = { vs[kk + 2 * hl][w * DHD + 16 + ll], vs[kk + 2 * hl + 1][w * DHD + 16 + ll] };
      o0 = wmma_f32_16x16x4(a, b0, o0);
      o1 = wmma_f32_16x16x4(a, b1, o1);
    }
  }
  __syncthreads();
  #pragma unroll
  for (int r = 0; r < 8; r++) {
    int rr = r + hl * 8;
    int e = q0 + rr;
    if (e < gend) {
      float inv = 1.f / rsum[w][rr];
      Oe[(size_t)e * HD + w * DHD + ll]      = o0[r] * inv;
      Oe[(size_t)e * HD + w * DHD + 16 + ll] = o1[r] * inv;
    }
  }
}

// ======================= temporal attention (per sequence, per head) =======================
__global__ void __launch_bounds__(128)
temporal_attn(const float* __restrict__ qkv, float* __restrict__ out) {
  __shared__ float qh[TTS][36];
  __shared__ float kh[TTS][36];
  __shared__ float vh[TTS][36];
  __shared__ float sm[TTS][68];
  int nb = blockIdx.x, hh = blockIdx.y;
  int tid = threadIdx.x;
  for (int i = tid; i < TTS * DHD; i += 128) {
    int r = i >> 5, d = i & 31;
    size_t rowoff = ((size_t)(nb * TTS + r)) * (3 * HD) + hh * DHD + d;
    qh[r][d] = qkv[rowoff];
    kh[r][d] = qkv[rowoff + HD];
    vh[r][d] = qkv[rowoff + 2 * HD];
  }
  __syncthreads();
  int w = tid >> 5, lane = tid & 31, ll = lane & 15, hl = lane >> 4;
  int m0 = w * 16;
  const float scale = 0.17677669529663687f;
  for (int n0 = 0; n0 < TTS; n0 += 16) {
    v8f acc = {};
    #pragma unroll
    for (int kk = 0; kk < DHD; kk += 4) {
      v2f a = { qh[m0 + ll][kk + 2 * hl], qh[m0 + ll][kk + 2 * hl + 1] };
      v2f b = { kh[n0 + ll][kk + 2 * hl], kh[n0 + ll][kk + 2 * hl + 1] };
      acc = wmma_f32_16x16x4(a, b, acc);
    }
    #pragma unroll
    for (int r = 0; r < 8; r++) sm[m0 + r + hl * 8][n0 + ll] = acc[r] * scale;
  }
  __syncthreads();
  if (tid < TTS) {
    float mx = -1e30f;
    for (int c = 0; c < TTS; c++) mx = fmaxf(mx, sm[tid][c]);
    float s = 0.f;
    for (int c = 0; c < TTS; c++) { float p = expf(sm[tid][c] - mx); sm[tid][c] = p; s += p; }
    float inv = 1.f / s;
    for (int c = 0; c < TTS; c++) sm[tid][c] *= inv;
  }
  __syncthreads();
  for (int d0 = 0; d0 < DHD; d0 += 16) {
    v8f acc = {};
    #pragma unroll
    for (int kk = 0; kk < TTS; kk += 4) {
      v2f a = { sm[m0 + ll][kk + 2 * hl], sm[m0 + ll][kk + 2 * hl + 1] };
      v2f b = { vh[kk + 2 * hl][d0 + ll], vh[kk + 2 * hl + 1][d0 + ll] };
      acc = wmma_f32_16x16x4(a, b, acc);
    }
    #pragma unroll
    for (int r = 0; r < 8; r++)
      out[((size_t)(nb * TTS + m0 + r + hl * 8)) * HD + hh * DHD + d0 + ll] = acc[r];
  }
}

// ======================= host orchestration =======================
extern "C" void kernel_launch(void* const* d_in, const int* in_sizes, int n_in,
                              void* d_out, int out_size, void* d_ws, size_t ws_size,
                              hipStream_t stream) {
  (void)in_sizes; (void)n_in; (void)out_size; (void)ws_size;
  const float* x       = (const float*)d_in[0];
  const int*   ei      = (const int*)d_in[1];
  const int*   etyp    = (const int*)d_in[2];
  const int*   ntyp    = (const int*)d_in[3];
  // d_in[4]=N_base, d_in[5]=T (fixed by constants)
  const float* Win     = (const float*)d_in[6];
  const float* bin     = (const float*)d_in[7];
  const float* s_in_w  = (const float*)d_in[8];
  const float* s_in_b  = (const float*)d_in[9];
  const float* s_out_w = (const float*)d_in[10];
  const float* s_out_b = (const float*)d_in[11];
  const float* ln_s_g  = (const float*)d_in[12];
  const float* ln_s_b  = (const float*)d_in[13];
  const float* t_in_w  = (const float*)d_in[14];
  const float* t_in_b  = (const float*)d_in[15];
  const float* t_out_w = (const float*)d_in[16];
  const float* t_out_b = (const float*)d_in[17];
  const float* t_ln1_g = (const float*)d_in[18];
  const float* t_ln1_b = (const float*)d_in[19];
  const float* t_w1    = (const float*)d_in[20];
  const float* t_b1    = (const float*)d_in[21];
  const float* t_w2    = (const float*)d_in[22];
  const float* t_b2    = (const float*)d_in[23];
  const float* t_ln2_g = (const float*)d_in[24];
  const float* t_ln2_b = (const float*)d_in[25];
  const float* fin_g   = (const float*)d_in[26];
  const float* fin_b   = (const float*)d_in[27];

  // workspace carve
  float* h    = (float*)d_ws;                        // NN*HD
  float* hnew = h + (size_t)NN * HD;                 // NN*HD (also temporal attn-out / ff-out)
  float* big  = hnew + (size_t)NN * HD;              // NN*384 floats shared scratch
  float* QKVe  = big;                                // spatial: EE*384
  float* eattn = big + (size_t)EE * 3 * HD;          // spatial: EE*128
  float* eout  = eattn + (size_t)EE * HD;            // spatial: EE*128
  float* qkvT  = big;                                // temporal: NN*384
  float* oproj = big;                                // temporal: NN*128 (reuse after attn)
  float* midb  = big;                                // temporal FFN chunk: 16384*512 (reuse)
  int* ip     = (int*)(big + (size_t)NN * 3 * HD);
  int* eperm  = ip;  ip += EE;
  int* idxQ   = ip;  ip += EE;
  int* idxKV  = ip;  ip += EE;
  int* goff   = ip;  ip += NGRP + 1;
  int* hist   = ip;  ip += NCHK * NGRP;
  int* gstart = ip;  ip += NCHK * NGRP;
  int* tileg  = ip;  ip += MAXTILES;
  int* tileq  = ip;  ip += MAXTILES;
  int* ntt    = ip;  ip += 1;

  // ---- deterministic edge grouping ----
  edge_hist<<<NCHK, CHK, 0, stream>>>(ei, etyp, ntyp, hist);
  edge_scan<<<1, 1, 0, stream>>>(hist, gstart, goff);
  edge_place<<<NCHK, 1, 0, stream>>>(ei, etyp, ntyp, gstart, eperm);
  build_idx<<<(EE + 255) / 256, 256, 0, stream>>>(ei, eperm, idxQ, idxKV);
  build_tiles<<<1, 1, 0, stream>>>(goff, tileg, tileq, ntt);

  // ---- input projection ----
  input_proj<<<NN, HD, 0, stream>>>(x, ntyp, Win, bin, h);

  // ---- spatial layers ----
  for (int li = 0; li < NLS; li++) {
    zero_f32<<<(NN * HD + 255) / 256, 256, 0, stream>>>(hnew, NN * HD);
    for (int et = 0; et < ETY; et++)
      spatial_qkv<<<dim3(EB64, 6), 128, 0, stream>>>(
          h, idxQ, idxKV, goff, et,
          s_in_w + (size_t)(li * ETY + et) * 3 * HD * HD,
          s_in_b + (size_t)(li * ETY + et) * 3 * HD, QKVe);
    spatial_attn<<<MAXTILES, 128, 0, stream>>>(QKVe, goff, tileg, tileq, ntt, eattn);
    for (int et = 0; et < ETY; et++)
      spatial_outproj<<<dim3(EB64, 2), 128, 0, stream>>>(
          eattn, goff, et,
          s_out_w + (size_t)(li * ETY + et) * HD * HD,
          s_out_b + (size_t)(li * ETY + et) * HD, eout);
    scatter_add<<<EE, HD, 0, stream>>>(eout, idxQ, hnew);
    ln_gelu_res<<<NN / 4, 128, 0, stream>>>(h, hnew, ln_s_g + li * HD, ln_s_b + li * HD,
                                            li > 0 ? 1 : 0);
  }

  // ---- temporal transformer ----
  pe_add<<<NN, HD, 0, stream>>>(h);
  for (int li = 0; li < NTL; li++) {
    gemm_plain<<<dim3(NN / 64, 6), 128, 0, stream>>>(
        h, HD, NN, t_in_w + (size_t)li * 3 * HD * HD, HD, t_in_b + li * 3 * HD, qkvT, 3 * HD, 0);
    temporal_attn<<<dim3(NBB, NHD), 128, 0, stream>>>(qkvT, hnew);
    gemm_plain<<<dim3(NN / 64, 2), 128, 0, stream>>>(
        hnew, HD, NN, t_out_w + (size_t)li * HD * HD, HD, t_out_b + li * HD, oproj, HD, 0);
    add_ln<<<NN / 4, 128, 0, stream>>>(h, oproj, t_ln1_g + li * HD, t_ln1_b + li * HD);
    for (int c = 0; c < 4; c++) {
      size_t r0 = (size_t)c * 16384;
      gemm_plain<<<dim3(16384 / 64, 8), 128, 0, stream>>>(
          h + r0 * HD, HD, 16384, t_w1 + (size_t)li * FFD * HD, HD, t_b1 + li * FFD,
          midb, FFD, 1 /*gelu*/);
      gemm_plain<<<dim3(16384 / 64, 2), 128, 0, stream>>>(
          midb, FFD, 16384, t_w2 + (size_t)li * HD * FFD, FFD, t_b2 + li * HD,
          hnew + r0 * HD, HD, 0);
    }
    add_ln<<<NN / 4, 128, 0, stream>>>(h, hnew, t_ln2_g + li * HD, t_ln2_b + li * HD);
  }
  final_ln<<<NN / 4, 128, 0, stream>>>(h, fin_g, fin_b, (float*)d_out);
}